// LogicDense_34368328302783
// MI455X (gfx1250) — compile-verified
//
#include <hip/hip_runtime.h>
#include <stdint.h>

// ---------------------------------------------------------------------------
// Problem constants (from reference)
// ---------------------------------------------------------------------------
#define BATCH   4096
#define IN_DIM  8192
#define OUT_DIM 16384

#define ROWS_PER_BLOCK 4          // batch rows staged in LDS per block
#define MAIN_THREADS   256
#define COEF_THREADS   256

typedef float        v2f   __attribute__((ext_vector_type(2)));
typedef float        v8f   __attribute__((ext_vector_type(8)));
typedef unsigned int u32x4 __attribute__((ext_vector_type(4)));
typedef int          i32x4 __attribute__((ext_vector_type(4)));
typedef int          i32x8 __attribute__((ext_vector_type(8)));

// GATE_COEFFS columns packed as nibbles (value + 2), nibble k at bit 4*k.
// Column c (0..3) gives coefficient GATE_COEFFS[k][c] for gate k.
#define GC_PACK_C0  0x3333333322222222ull   // c0
#define GC_PACK_CA  0x2211221133223322ull   // ca
#define GC_PACK_CB  0x2222111133332222ull   // cb
#define GC_PACK_CAB 0x2132324310212132ull   // cab

// ---------------------------------------------------------------------------
// Kernel 1: coeffs[j][0..3] = softmax(weight[j,:]) @ GATE_COEFFS
// One wave handles 16 rows of `weight` via V_WMMA_F32_16X16X4_F32.
//
// A-matrix 16x4 f32 layout (ISA 7.12.2): lanes 0-15 hold M=0..15 with
// K=0 (v0) / K=1 (v1); lanes 16-31 hold K=2 (v0) / K=3 (v1).
// For K-chunk p, lane (half h = lane>>4, m = lane&15) supplies
// w_sm[row m][4p + 2h + {0,1}].
// B-matrix 4x16 f32: v0 = chunk row (2h), v1 = chunk row (2h+1),
// column index = m; columns >= 4 are the zero padding. B entries are
// synthesized from the nibble-packed constants (pure VALU, no loads).
// ---------------------------------------------------------------------------
__global__ void __launch_bounds__(COEF_THREADS)
logic_coeffs_wmma(const float* __restrict__ weight, float* __restrict__ coeffs)
{
    const int tid  = blockIdx.x * blockDim.x + threadIdx.x;
    const int lane = threadIdx.x & 31;
    const int wave = tid >> 5;            // 1024 waves total
    const int rowbase = wave << 4;        // 16 weight rows per wave
    const int m    = lane & 15;           // A: row index / B,D: column index
    const int half = lane >> 4;
    const int r    = rowbase + m;

    // This lane's packed GATE_COEFFS column (branchless select on m&3; m>=4 zeroed in B below).
    uint64_t pm = (m & 2) ? ((m & 1) ? GC_PACK_CAB : GC_PACK_CB)
                          : ((m & 1) ? GC_PACK_CA  : GC_PACK_C0);

    // Load this lane's 8 of 16 weight entries for row r (4 chunks x 2 K's).
    const float* wr = weight + (size_t)r * 16 + 2 * half;
    float v[4][2];
#pragma unroll
    for (int p = 0; p < 4; ++p) {
        float2 t = *(const float2*)(wr + 4 * p);   // offsets are even -> 8B aligned
        v[p][0] = t.x;
        v[p][1] = t.y;
    }

    // Softmax over the 16 entries of row r (TEMPERATURE == 1).
    // Partial over this lane's 8, combined with the partner lane (xor 16).
    float mx = -3.402823466e38f;
#pragma unroll
    for (int p = 0; p < 4; ++p) { mx = fmaxf(mx, fmaxf(v[p][0], v[p][1])); }
    mx = fmaxf(mx, __shfl_xor(mx, 16, 32));

    float e[4][2];
    float s = 0.0f;
#pragma unroll
    for (int p = 0; p < 4; ++p) {
        e[p][0] = expf(v[p][0] - mx);
        e[p][1] = expf(v[p][1] - mx);
        s += e[p][0] + e[p][1];
    }
    s += __shfl_xor(s, 16, 32);
    const float inv = __builtin_amdgcn_rcpf(s);    // v_rcp_f32

    // 4 accumulating WMMAs over K=16 against zero-padded 16x16 GATE_COEFFS.
    v8f acc = {};
#pragma unroll
    for (int p = 0; p < 4; ++p) {
        v2f a;
        a.x = e[p][0] * inv;
        a.y = e[p][1] * inv;
        const int k0 = 4 * p + 2 * half;
        const float b0 = (float)((int)((pm >> (4 * k0))       & 0xFull) - 2);
        const float b1 = (float)((int)((pm >> (4 * (k0 + 1))) & 0xFull) - 2);
        v2f b;
        b.x = (m < 4) ? b0 : 0.0f;
        b.y = (m < 4) ? b1 : 0.0f;
        acc = __builtin_amdgcn_wmma_f32_16x16x4_f32(
            /*neg_a=*/false, a, /*neg_b=*/false, b,
            /*c_mod=*/(short)0, acc, /*reuse_a=*/false, /*reuse_b=*/false);
    }

    // D 16x16 f32 layout: VGPR vv, lanes 0-15 -> M=vv, N=lane;
    // lanes 16-31 -> M=vv+8, N=lane-16. Only N<4 is meaningful.
    if (m < 4) {
#pragma unroll
        for (int vv = 0; vv < 8; ++vv) {
            const int row = rowbase + vv + 8 * half;
            coeffs[(size_t)row * 4 + m] = acc[vv];
        }
    }
}

// ---------------------------------------------------------------------------
// Kernel 2: main gather + elementwise.
// Block b stages batch rows [b*4, b*4+4) of x (128 KB) into LDS with one
// Tensor Data Mover descriptor, then sweeps all 16384 output columns.
// x is read from HBM exactly once, coalesced; gathers are LDS-local;
// stores are coalesced along j.
// ---------------------------------------------------------------------------
__global__ void __launch_bounds__(MAIN_THREADS)
logic_dense_main(const float* __restrict__ x,
                 const int*   __restrict__ indices,   // [2][OUT_DIM]
                 const float* __restrict__ coeffs,    // [OUT_DIM][4]
                 float*       __restrict__ out)       // [BATCH][OUT_DIM]
{
    extern __shared__ float lds[];                    // ROWS_PER_BLOCK * IN_DIM floats
    const int i0 = blockIdx.x * ROWS_PER_BLOCK;

    if (threadIdx.x == 0) {
        // --- Tensor DMA descriptor (D#): 2D tile, 4 rows x 8192 f32 ---
        const uint64_t ga = (uint64_t)(uintptr_t)(x + (size_t)i0 * IN_DIM);
        const uint32_t lds_off = (uint32_t)(uintptr_t)(void*)lds; // low 32 bits = LDS byte offset

        u32x4 g0;
        g0.x = 1u;                                          // count = 1 (valid descriptor)
        g0.y = lds_off;                                     // lds_addr
        g0.z = (uint32_t)ga;                                // global_addr[31:0]  (bits 95:64)
        g0.w = (uint32_t)((ga >> 32) & 0x1FFFFFFu)          // global_addr[56:32]
             | (2u << 30);                                  // type = 2 ("image")

        i32x8 g1;
        g1[0] = 0x00020000;                 // workgroup_mask=0, data_size=2 (4 bytes)
        g1[1] = (int)(IN_DIM << 16);        // tensor_dim0[15:0] = 8192 @ bits 63:48
        g1[2] = (int)(ROWS_PER_BLOCK << 16);// tensor_dim0[31:16]=0 | tensor_dim1[15:0]=4
        g1[3] = (int)(IN_DIM << 16);        // tensor_dim1[31:16]=0 | tile_dim0 = 8192
        g1[4] = ROWS_PER_BLOCK;             // tile_dim1 = 4, tile_dim2 = 0
        g1[5] = IN_DIM;                     // tensor_dim0_stride[31:0] = 8192
        g1[6] = 0;                          // tensor_dim0_stride[47:32], dim1_stride lo
        g1[7] = 0;

        i32x4 gz4 = {};                     // groups 2/3 unused (<= 2D tensor)
        i32x8 gz8 = {};                     // 6-arg form: extra group (unused)
        __builtin_amdgcn_tensor_load_to_lds(g0, g1, gz4, gz4, gz8, 0);
        __builtin_amdgcn_s_wait_tensorcnt(0);
    }
    __syncthreads();

    const int* idx0 = indices;
    const int* idx1 = indices + OUT_DIM;

    for (int j = threadIdx.x; j < OUT_DIM; j += MAIN_THREADS) {
        const int ia = idx0[j];
        const int ib = idx1[j];
        const float4 c = ((const float4*)coeffs)[j];   // c0, ca, cb, cab
#pragma unroll
        for (int r = 0; r < ROWS_PER_BLOCK; ++r) {
            const float a = lds[r * IN_DIM + ia];
            const float b = lds[r * IN_DIM + ib];
            const float res = fmaf(c.w, a * b, fmaf(c.z, b, fmaf(c.y, a, c.x)));
            out[(size_t)(i0 + r) * OUT_DIM + j] = res;
        }
    }
}

// ---------------------------------------------------------------------------
// Launch
// ---------------------------------------------------------------------------
extern "C" void kernel_launch(void* const* d_in, const int* in_sizes, int n_in,
                              void* d_out, int out_size, void* d_ws, size_t ws_size,
                              hipStream_t stream)
{
    const float* x       = (const float*)d_in[0];   // [BATCH, IN_DIM] f32
    const float* weight  = (const float*)d_in[1];   // [OUT_DIM, 16]  f32
    const int*   indices = (const int*)d_in[2];     // [2, OUT_DIM]   i32
    float*       out     = (float*)d_out;           // [BATCH, OUT_DIM] f32
    float*       coeffs  = (float*)d_ws;            // [OUT_DIM, 4] scratch (256 KB)

    // Kernel 1: 1024 waves (16 rows each) -> 32768 threads.
    {
        const int total_threads = (OUT_DIM / 16) * 32;
        const int blocks = total_threads / COEF_THREADS;
        logic_coeffs_wmma<<<blocks, COEF_THREADS, 0, stream>>>(weight, coeffs);
    }

    // Kernel 2: one block per 4 batch rows, 128 KB dynamic LDS.
    {
        const int blocks = BATCH / ROWS_PER_BLOCK;  // 1024
        const size_t shmem = (size_t)ROWS_PER_BLOCK * IN_DIM * sizeof(float); // 131072
        logic_dense_main<<<blocks, MAIN_THREADS, shmem, stream>>>(x, indices, coeffs, out);
    }
}